// Hamming_34205119545967
// MI455X (gfx1250) — compile-verified
//
#include <hip/hip_runtime.h>
#include <hip/hip_bf16.h>
#include <stdint.h>

// Census Hamming-distance cost volume (stereo), MI455X / gfx1250.
// Reference shapes: B=2, C=1, H=384, W=768, D=128, 24-bit census values.
// out[b,d,h,x] = (x+d < W) ? popcount(img1[b,h,x+d] ^ img2[b,h,x]) * scale : 0
//
// Store-bound: 302 MB of output vs 4.7 MB of input -> stage each row pair in
// LDS once (async global->LDS), reuse 128x, stream results out with NT stores.
// HBM floor at 23.3 TB/s: ~13 us. WMMA inapplicable (Toeplitz correlation:
// the left index x+d depends on both output coords, so no A[m,k]*B[k,n] form).

#define BATCH 2
#define HGT   384
#define WID   768
#define DISP  128
#define PADW  (WID + DISP)   // left row zero-padded so sliding window never reads OOB
#define TPB   192            // 6 waves; 4 x-columns per thread (4*192 == W)

typedef float vf4 __attribute__((ext_vector_type(4)));
typedef int   v4i __attribute__((ext_vector_type(4)));
typedef __attribute__((address_space(1))) v4i* gptr_v4i;   // global
typedef __attribute__((address_space(3))) v4i* lptr_v4i;   // LDS

#if defined(__gfx1250__) && __has_builtin(__builtin_amdgcn_global_load_async_to_lds_b128)
#define USE_ASYNC 1
#else
#define USE_ASYNC 0
#endif

__global__ __launch_bounds__(TPB) void census_hamming_kernel(
    const int* __restrict__ img1,
    const int* __restrict__ img2,
    const float* __restrict__ scale,
    float* __restrict__ out)
{
    __shared__ int ldsL[PADW];
    __shared__ int ldsR[WID];

    const int h = blockIdx.x;
    const int b = blockIdx.y;
    const int t = threadIdx.x;

    const int* gL = img1 + ((size_t)b * HGT + h) * WID;
    const int* gR = img2 + ((size_t)b * HGT + h) * WID;

    // Zero-fill the padded tail of the left row (DISP entries; threads 0..127).
    if (t < PADW - WID) ldsL[WID + t] = 0;

#if USE_ASYNC
    {
        // One async B128 transfer per thread per image: 192 lanes * 16 B = full 3 KB row.
        __builtin_amdgcn_global_load_async_to_lds_b128(
            (gptr_v4i)(gL + 4 * t), (lptr_v4i)(ldsL + 4 * t), 0, 0);
        __builtin_amdgcn_global_load_async_to_lds_b128(
            (gptr_v4i)(gR + 4 * t), (lptr_v4i)(ldsR + 4 * t), 0, 0);
#if __has_builtin(__builtin_amdgcn_s_wait_asynccnt)
        __builtin_amdgcn_s_wait_asynccnt(0);
#else
        asm volatile("s_wait_asynccnt 0" ::: "memory");
#endif
    }
#else
    for (int i = t; i < WID; i += TPB) { ldsL[i] = gL[i]; ldsR[i] = gR[i]; }
#endif
    __syncthreads();

    const float s  = scale[0];
    const int   x0 = 4 * t;

    // Right-image words: loop-invariant, one aligned 16B LDS read, kept in VGPRs.
    const int r0 = ldsR[x0 + 0];
    const int r1 = ldsR[x0 + 1];
    const int r2 = ldsR[x0 + 2];
    const int r3 = ldsR[x0 + 3];

    // Sliding left-image window: L[x0+d .. x0+d+3]; shifts by one per disparity.
    int l0 = ldsL[x0 + 0];
    int l1 = ldsL[x0 + 1];
    int l2 = ldsL[x0 + 2];
    int l3 = ldsL[x0 + 3];

    float* orow = out + ((size_t)b * DISP * HGT + h) * WID + x0;
    const size_t dstride = (size_t)HGT * WID;   // floats between consecutive d-planes

#pragma unroll 4
    for (int d = 0; d < DISP; ++d) {
        const int lim = WID - d;   // x valid iff x < lim  (i.e. x + d < W)
        const int h0 = (x0 + 0 < lim) ? __popc(l0 ^ r0) : 0;
        const int h1 = (x0 + 1 < lim) ? __popc(l1 ^ r1) : 0;
        const int h2 = (x0 + 2 < lim) ? __popc(l2 ^ r2) : 0;
        const int h3 = (x0 + 3 < lim) ? __popc(l3 ^ r3) : 0;

        vf4 v;
        v.x = (float)h0 * s;
        v.y = (float)h1 * s;
        v.z = (float)h2 * s;
        v.w = (float)h3 * s;
        // Streaming 16B store; 302 MB total output should not wash L2.
        __builtin_nontemporal_store(v, (vf4*)orow);
        orow += dstride;

        // Shift window; one new LDS dword per disparity. Max index = 764+4+127 = 895 < PADW.
        l0 = l1; l1 = l2; l2 = l3;
        l3 = ldsL[x0 + 4 + d];
    }
}

extern "C" void kernel_launch(void* const* d_in, const int* in_sizes, int n_in,
                              void* d_out, int out_size, void* d_ws, size_t ws_size,
                              hipStream_t stream) {
    const int*   img1  = (const int*)d_in[0];
    const int*   img2  = (const int*)d_in[1];
    const float* scale = (const float*)d_in[2];
    float*       out   = (float*)d_out;

    dim3 grid(HGT, BATCH, 1);   // one workgroup per (row, batch)
    census_hamming_kernel<<<grid, dim3(TPB, 1, 1), 0, stream>>>(img1, img2, scale, out);
}